// Filter_42331197670043
// MI455X (gfx1250) — compile-verified
//
#include <hip/hip_runtime.h>
#include <hip/hip_bf16.h>
#include <stdint.h>

// ---------------------------------------------------------------------------
// FilterDetections (per-class greedy NMS + global top-k) for MI455X / gfx1250.
//
// Stage 1: one 1024-thread workgroup per (batch, class) pair (160 blocks).
//   - TDM (tensor_load_to_lds) DMAs the whole [N,4] f32 box tensor (320000 B)
//     of this batch into LDS (CDNA5: 320KB LDS per workgroup).
//   - Per-thread: 20 class scores in VGPRs (thresholded to -inf).
//   - 300 greedy iterations: block argmax (wave32 shfl_xor + LDS tree),
//     broadcast winner box from LDS, IoU-suppress owned candidates.
//   - Emits per-(b,c): 300 winner scores (-inf if invalid) + anchor indices.
// Stage 2: one workgroup per batch.
//   - TDM loads the 80*300 candidate scores (96000 B) into LDS.
//   - 300 selection rounds (value desc, index asc == jax.lax.top_k order),
//     gathers boxes/labels, writes outputs (invalid -> -1).
// ---------------------------------------------------------------------------

#define BATCH     2
#define N_ANCH    20000
#define NCLS      80
#define MAXDET    300
#define K1        20            // ceil(N_ANCH / 1024)
#define NCAND     (NCLS * MAXDET)   // 24000
#define K2        24            // ceil(NCAND / 1024)

__device__ __forceinline__ float neg_inf() { return -__builtin_inff(); }

typedef unsigned int u32x4 __attribute__((ext_vector_type(4)));
typedef int          i32x4 __attribute__((ext_vector_type(4)));
typedef int          i32x8 __attribute__((ext_vector_type(8)));

// ---- Tensor Data Mover: 2D f32 tile (dim0 x dim1, row stride stride0),
// global -> LDS at byte offset lds_addr. Caller must be a single thread,
// then s_wait_tensorcnt(0) + __syncthreads() before LDS is consumed.
__device__ __forceinline__ void tdm_load_f32_2d(const void* gptr,
                                                uint32_t lds_addr,
                                                uint32_t dim0, uint32_t dim1,
                                                uint64_t stride0) {
  uint64_t ga = (uint64_t)gptr;
  u32x4 g0;
  g0.x = 1u;                                            // count=1 (valid D#)
  g0.y = lds_addr;                                      // LDS byte address
  g0.z = (uint32_t)ga;                                  // global_addr[31:0]
  g0.w = (uint32_t)((ga >> 32) & 0x1FFFFFFu) | (2u << 30); // addr[56:32]|type=2

  uint64_t s1 = stride0 * (uint64_t)dim1;               // dim1 stride (elts)
  i32x8 g1;
  g1[0] = (int)(2u << 16);                              // data_size = 4B
  g1[1] = (int)((dim0 & 0xFFFFu) << 16);                // tensor_dim0 lo16
  g1[2] = (int)(((dim1 & 0xFFFFu) << 16) | ((dim0 >> 16) & 0xFFFFu));
  g1[3] = (int)(((dim0 & 0xFFFFu) << 16) | ((dim1 >> 16) & 0xFFFFu)); // tile_dim0 | tdim1 hi
  g1[4] = (int)(dim1 & 0xFFFFu);                        // tile_dim1 (tile_dim2=0)
  g1[5] = (int)(uint32_t)(stride0 & 0xFFFFFFFFu);       // tdim0_stride lo32
  g1[6] = (int)(((uint32_t)(s1 & 0xFFFFu) << 16) |
                (uint32_t)((stride0 >> 32) & 0xFFFFu)); // tdim1_stride lo16 | tdim0_stride hi16
  g1[7] = (int)(uint32_t)((s1 >> 16) & 0xFFFFFFFFu);    // tdim1_stride[47:16]

  i32x4 g2 = {0, 0, 0, 0};
  i32x4 g3 = {0, 0, 0, 0};
#if defined(__clang_major__) && (__clang_major__ >= 23)
  i32x8 g4 = {0, 0, 0, 0, 0, 0, 0, 0};
  __builtin_amdgcn_tensor_load_to_lds(g0, g1, g2, g3, g4, 0);
#else
  __builtin_amdgcn_tensor_load_to_lds(g0, g1, g2, g3, 0);
#endif
}

// ---- wave32 argmax reduction: (score desc, index asc) ----
__device__ __forceinline__ void wave_argmax(float& s, int& i) {
#pragma unroll
  for (int m = 16; m >= 1; m >>= 1) {
    float s2 = __shfl_xor(s, m, 32);
    int   i2 = __shfl_xor(i, m, 32);
    if (s2 > s || (s2 == s && i2 < i)) { s = s2; i = i2; }
  }
}

// Block argmax across 32 waves (1024 threads). Two barriers per call.
__device__ __forceinline__ void block_argmax(float& s, int& i, int tid,
                                             float* rs, int* ri,
                                             float* win_s, int* win_i) {
  wave_argmax(s, i);
  int w = tid >> 5, lane = tid & 31;
  if (lane == 0) { rs[w] = s; ri[w] = i; }
  __syncthreads();
  if (w == 0) {
    float s0 = rs[lane];
    int   i0 = ri[lane];
    wave_argmax(s0, i0);
    if (lane == 0) { *win_s = s0; *win_i = i0; }
  }
  __syncthreads();
  s = *win_s; i = *win_i;
}

// ===========================================================================
// Stage 1: per-(batch,class) greedy NMS
// ===========================================================================
__global__ __launch_bounds__(1024)
void nms_stage1(const float* __restrict__ boxes,
                const float* __restrict__ cls,
                float* __restrict__ ws_sc,   // [B][C][MAXDET]
                int*   __restrict__ ws_ix) { // [B][C][MAXDET]
  __shared__ float4 lds_boxes[N_ANCH];       // 320000 B (CDNA5 320KB LDS)
  __shared__ float  rs[32];
  __shared__ int    ri[32];
  __shared__ float  win_s;
  __shared__ int    win_i;

  const int tid = threadIdx.x;
  const int b   = blockIdx.x / NCLS;
  const int c   = blockIdx.x % NCLS;
  const float NI = neg_inf();

  // --- TDM: DMA this batch's boxes [N_ANCH,4] f32 into LDS, one issue/WG ---
  if (tid == 0) {
    const void* g = (const void*)(boxes + (size_t)b * N_ANCH * 4);
    uint32_t lds_off = (uint32_t)(uintptr_t)(void*)&lds_boxes[0];
    tdm_load_f32_2d(g, lds_off, /*dim0=*/2000, /*dim1=*/40, /*stride0=*/2000);
    __builtin_amdgcn_s_wait_tensorcnt(0);
  }
  __syncthreads();

  // --- load + threshold this class's scores into registers ---
  float sc[K1];
#pragma unroll
  for (int k = 0; k < K1; ++k) {
    int i = tid + (k << 10);
    float v = (i < N_ANCH) ? cls[((size_t)b * N_ANCH + i) * NCLS + c] : 0.0f;
    sc[k] = (i < N_ANCH && v > 0.05f) ? v : NI;
  }

  const size_t base = ((size_t)b * NCLS + c) * MAXDET;

  for (int t = 0; t < MAXDET; ++t) {
    // local argmax (ascending index ownership => strict '>' keeps first max)
    float bs = NI; int bi = 0;
#pragma unroll
    for (int k = 0; k < K1; ++k) {
      int i = tid + (k << 10);
      if (sc[k] > bs) { bs = sc[k]; bi = i; }
    }
    block_argmax(bs, bi, tid, rs, ri, &win_s, &win_i);

    if (tid == 0) { ws_sc[base + t] = bs; ws_ix[base + t] = bi; }

    if (!(bs > NI)) {
      // nothing above threshold remains: all further slots invalid
      for (int t2 = t + tid; t2 < MAXDET; t2 += 1024) {
        ws_sc[base + t2] = NI;
        ws_ix[base + t2] = 0;
      }
      break;
    }

    // winner box broadcast from LDS, IoU suppression of owned candidates
    float4 wb = lds_boxes[bi];
    float warea = (wb.z - wb.x) * (wb.w - wb.y);
#pragma unroll
    for (int k = 0; k < K1; ++k) {
      int i = tid + (k << 10);
      if (i < N_ANCH && sc[k] > NI) {
        float4 bx = lds_boxes[i];
        float xx1 = fmaxf(bx.x, wb.x);
        float yy1 = fmaxf(bx.y, wb.y);
        float xx2 = fminf(bx.z, wb.z);
        float yy2 = fminf(bx.w, wb.w);
        float inter = fmaxf(xx2 - xx1, 0.0f) * fmaxf(yy2 - yy1, 0.0f);
        float a = (bx.z - bx.x) * (bx.w - bx.y);
        float iou = inter / (a + warea - inter);
        if (iou > 0.5f || i == bi) sc[k] = NI;
      }
    }
  }
}

// ===========================================================================
// Stage 2: per-batch top-300 over the 80*300 candidates
// ===========================================================================
__global__ __launch_bounds__(1024)
void nms_stage2(const float* __restrict__ boxes,
                const float* __restrict__ ws_sc,  // [B][NCAND]
                const int*   __restrict__ ws_ix,  // [B][NCAND]
                float* __restrict__ out) {
  __shared__ float lds_sc[NCAND];              // 96000 B
  __shared__ float rs[32];
  __shared__ int   ri[32];
  __shared__ float win_s;
  __shared__ int   win_i;

  const int tid = threadIdx.x;
  const int b   = blockIdx.x;
  const float NI = neg_inf();

  // --- TDM: DMA this batch's candidate scores into LDS ---
  if (tid == 0) {
    const void* g = (const void*)(ws_sc + (size_t)b * NCAND);
    uint32_t lds_off = (uint32_t)(uintptr_t)(void*)&lds_sc[0];
    tdm_load_f32_2d(g, lds_off, /*dim0=*/2400, /*dim1=*/10, /*stride0=*/2400);
    __builtin_amdgcn_s_wait_tensorcnt(0);
  }
  __syncthreads();

  float cs[K2];
#pragma unroll
  for (int k = 0; k < K2; ++k) {
    int f = tid + (k << 10);
    cs[k] = (f < NCAND) ? lds_sc[f] : NI;
  }

  // output regions (flat, return order): boxes[2][300][4], scores, labels
  const int OUT_SCORES = BATCH * MAXDET * 4;          // 2400
  const int OUT_LABELS = OUT_SCORES + BATCH * MAXDET; // 3000

  for (int t = 0; t < MAXDET; ++t) {
    float bs = NI; int bf = 0;
#pragma unroll
    for (int k = 0; k < K2; ++k) {
      int f = tid + (k << 10);
      if (cs[k] > bs) { bs = cs[k]; bf = f; }
    }
    block_argmax(bs, bf, tid, rs, ri, &win_s, &win_i);

    if (!(bs > NI)) {
      for (int t2 = t + tid; t2 < MAXDET; t2 += 1024) {
        int o = (b * MAXDET + t2);
        out[o * 4 + 0] = -1.0f; out[o * 4 + 1] = -1.0f;
        out[o * 4 + 2] = -1.0f; out[o * 4 + 3] = -1.0f;
        out[OUT_SCORES + o] = -1.0f;
        out[OUT_LABELS + o] = -1.0f;
      }
      break;
    }

    if (tid == 0) {
      int c      = bf / MAXDET;                       // label = pos // 300
      int anchor = ws_ix[(size_t)b * NCAND + bf];
      const float* bb = boxes + ((size_t)b * N_ANCH + anchor) * 4;
      int o = b * MAXDET + t;
      out[o * 4 + 0] = bb[0]; out[o * 4 + 1] = bb[1];
      out[o * 4 + 2] = bb[2]; out[o * 4 + 3] = bb[3];
      out[OUT_SCORES + o] = bs;
      out[OUT_LABELS + o] = (float)c;
    }

    // remove the winner from its owner's registers (fully unrolled, no
    // dynamic register indexing -> stays in VGPRs)
    if (tid == (bf & 1023)) {
      int kk = bf >> 10;
#pragma unroll
      for (int k = 0; k < K2; ++k)
        if (k == kk) cs[k] = NI;
    }
  }
}

extern "C" void kernel_launch(void* const* d_in, const int* in_sizes, int n_in,
                              void* d_out, int out_size, void* d_ws, size_t ws_size,
                              hipStream_t stream) {
  const float* boxes = (const float*)d_in[0];          // [2, 20000, 4] f32
  const float* cls   = (const float*)d_in[1];          // [2, 20000, 80] f32
  float* out = (float*)d_out;                          // 3600 f32

  // workspace: [B][C][300] scores (f32) then [B][C][300] anchor idx (i32)
  float* ws_sc = (float*)d_ws;
  int*   ws_ix = (int*)((char*)d_ws + (size_t)BATCH * NCLS * MAXDET * 4);

  nms_stage1<<<dim3(BATCH * NCLS), dim3(1024), 0, stream>>>(boxes, cls, ws_sc, ws_ix);
  nms_stage2<<<dim3(BATCH), dim3(1024), 0, stream>>>(boxes, ws_sc, ws_ix, out);
}